// RegionLossV3_72851235275378
// MI455X (gfx1250) — compile-verified
//
#include <hip/hip_runtime.h>
#include <hip/hip_bf16.h>
#include <math.h>

typedef __attribute__((ext_vector_type(2))) float v2f;
typedef __attribute__((ext_vector_type(8))) float v8f;

#define NAANCH 5
#define NH 26
#define NW 26
#define HW 676
#define KPOS 3380      // NA*NH*NW
#define NB 320
#define CSC 20         // CS (class/tile dim)
#define MAXB 50
#define NMETA 20
#define EPAD 32        // enews padded rows for WMMA
#define DW_N 64
#define DW_D 1024

__constant__ float c_aw[NAANCH] = {1.3221f, 3.19275f, 5.05587f, 9.47112f, 11.2364f};
__constant__ float c_ah[NAANCH] = {1.73145f, 4.00944f, 8.09892f, 4.84053f, 10.0071f};

__device__ __forceinline__ float sigm(float v) { return 1.0f / (1.0f + expf(-v)); }

__device__ __forceinline__ float iou8(float cx1, float cy1, float w1, float h1,
                                      float cx2, float cy2, float w2, float h2) {
    float mx = fminf(cx1 - 0.5f * w1, cx2 - 0.5f * w2);
    float Mx = fmaxf(cx1 + 0.5f * w1, cx2 + 0.5f * w2);
    float my = fminf(cy1 - 0.5f * h1, cy2 - 0.5f * h2);
    float My = fmaxf(cy1 + 0.5f * h1, cy2 + 0.5f * h2);
    float cw = w1 + w2 - (Mx - mx);
    float ch = h1 + h2 - (My - my);
    float inter = (cw > 0.0f && ch > 0.0f) ? cw * ch : 0.0f;
    return inter / (w1 * h1 + w2 * h2 - inter);
}

// ---------------------------------------------------------------------------
// Kernel 1: base losses over every (b, anchor, j, i) position.
// Invalid GT boxes are stored with a far-away sentinel center so their IoU is
// exactly 0 (same as where(valid, iou, 0) before the max) -> branch-free loop.
// ---------------------------------------------------------------------------
__global__ __launch_bounds__(256) void k_base(const float* __restrict__ out,
                                              const float* __restrict__ tgt,
                                              float* __restrict__ P1) {
    __shared__ float sgx[MAXB], sgy[MAXB], sgw[MAXB], sgh[MAXB];
    __shared__ float sred[256];
    const int b   = blockIdx.x;
    const int tid = threadIdx.x;
    if (tid < MAXB) {
        const float* g = tgt + b * 250 + tid * 5;
        float r1 = g[1], r2 = g[2], r3 = g[3], r4 = g[4];
        bool valid = (r1 != 0.0f);
        sgx[tid] = valid ? r1 * (float)NW : 1.0e8f;   // sentinel: no overlap -> IoU 0
        sgy[tid] = valid ? r2 * (float)NH : 1.0e8f;
        sgw[tid] = valid ? r3 * (float)NW : 1.0f;
        sgh[tid] = valid ? r4 * (float)NH : 1.0f;
    }
    __syncthreads();

    float loss = 0.0f;
    const int k = blockIdx.y * 256 + tid;
    if (k < KPOS) {
        const int a = k / HW;
        const int r = k - a * HW;
        const int j = r / NW;
        const int i = r - j * NW;
        const float* o = out + ((size_t)b * 30 + (size_t)a * 6) * HW + j * NW + i;
        const float ox = o[0];
        const float oy = o[HW];
        const float ow = o[2 * HW];
        const float oh = o[3 * HW];
        const float oc = o[4 * HW];
        const float x    = sigm(ox);
        const float y    = sigm(oy);
        const float conf = sigm(oc);
        const float px = x + (float)i;
        const float py = y + (float)j;
        const float pw = expf(ow) * c_aw[a];
        const float ph = expf(oh) * c_ah[a];
        float cur = 0.0f;
        #pragma unroll 5
        for (int t = 0; t < MAXB; ++t)
            cur = fmaxf(cur, iou8(px, py, pw, ph, sgx[t], sgy[t], sgw[t], sgh[t]));
        loss = 0.5f * ((x - 0.5f) * (x - 0.5f) + (y - 0.5f) * (y - 0.5f) + ow * ow + oh * oh);
        if (cur <= 0.6f) loss += 0.5f * conf * conf;  // NOOBJ_SCALE=1, tconf=0
    }

    sred[tid] = loss;
    __syncthreads();
    for (int s = 128; s > 0; s >>= 1) {
        if (tid < s) sred[tid] += sred[tid + s];
        __syncthreads();
    }
    if (tid == 0) P1[blockIdx.x * gridDim.y + blockIdx.y] = sred[0];
}

// ---------------------------------------------------------------------------
// Kernel 2: scattered-target corrections (one block per image, <=50 boxes).
// Duplicate fidx within an image: last-write-wins (matches set semantics).
// ---------------------------------------------------------------------------
__global__ __launch_bounds__(64) void k_scatter(const float* __restrict__ out,
                                                const float* __restrict__ tgt,
                                                float* __restrict__ P2) {
    __shared__ float sgx[MAXB], sgy[MAXB], sgw[MAXB], sgh[MAXB];
    __shared__ int   sval[MAXB], skey[MAXB], sbn[MAXB], sgiA[MAXB], sgjA[MAXB];
    __shared__ float sred[64];
    const int b   = blockIdx.x;
    const int tid = threadIdx.x;

    if (tid < MAXB) {
        const float* g = tgt + b * 250 + tid * 5;
        float r1 = g[1], r2 = g[2], r3 = g[3], r4 = g[4];
        int valid = (r1 != 0.0f) ? 1 : 0;
        float gx = r1 * (float)NW, gy = r2 * (float)NH;
        float gw = r3 * (float)NW, gh = r4 * (float)NH;
        sval[tid] = valid;
        sgx[tid] = valid ? gx : 1.0e8f;   // sentinel for branch-free max-IoU loop
        sgy[tid] = valid ? gy : 1.0e8f;
        sgw[tid] = valid ? gw : 1.0f;
        sgh[tid] = valid ? gh : 1.0f;
        int   bn   = 0;
        float best = -1.0f;
        #pragma unroll
        for (int a = 0; a < NAANCH; ++a) {
            float inter = fminf(gw, c_aw[a]) * fminf(gh, c_ah[a]);
            float u     = gw * gh + c_aw[a] * c_ah[a] - inter;
            float v     = inter / u;
            if (v > best) { best = v; bn = a; }   // first-max wins (argmax)
        }
        int gi = (int)gx; gi = gi < 0 ? 0 : (gi > NW - 1 ? NW - 1 : gi);
        int gj = (int)gy; gj = gj < 0 ? 0 : (gj > NH - 1 ? NH - 1 : gj);
        sbn[tid] = bn; sgiA[tid] = gi; sgjA[tid] = gj;
        skey[tid] = (bn * NH + gj) * NW + gi;
    }
    __syncthreads();

    float delta = 0.0f;
    if (tid < MAXB && sval[tid]) {
        bool keep = true;
        const int key = skey[tid];
        for (int t2 = tid + 1; t2 < MAXB; ++t2)
            if (sval[t2] && skey[t2] == key) { keep = false; break; }
        if (keep) {
            const int   bn = sbn[tid], gi = sgiA[tid], gj = sgjA[tid];
            const float gx = sgx[tid], gy = sgy[tid], gw = sgw[tid], gh = sgh[tid];
            const float* o = out + ((size_t)b * 30 + (size_t)bn * 6) * HW + gj * NW + gi;
            const float ox = o[0], oy = o[HW], ow = o[2 * HW], oh = o[3 * HW], oc = o[4 * HW];
            const float x = sigm(ox), y = sigm(oy), conf = sigm(oc);
            const float px = x + (float)gi, py = y + (float)gj;
            const float pw = expf(ow) * c_aw[bn], ph = expf(oh) * c_ah[bn];
            const float iou_gt = iou8(gx, gy, gw, gh, px, py, pw, ph);
            float cur = 0.0f;
            for (int t = 0; t < MAXB; ++t)
                cur = fmaxf(cur, iou8(px, py, pw, ph, sgx[t], sgy[t], sgw[t], sgh[t]));
            const float basemask = (cur > 0.6f) ? 0.0f : 1.0f;
            const float txn = gx - (float)gi, tyn = gy - (float)gj;
            const float twn = logf(gw / c_aw[bn]), thn = logf(gh / c_ah[bn]);
            delta += 0.5f * ((x - txn) * (x - txn) - (x - 0.5f) * (x - 0.5f));
            delta += 0.5f * ((y - tyn) * (y - tyn) - (y - 0.5f) * (y - 0.5f));
            delta += 0.5f * ((ow - twn) * (ow - twn) - ow * ow);
            delta += 0.5f * ((oh - thn) * (oh - thn) - oh * oh);
            delta += 0.5f * (5.0f * (conf - iou_gt) * (conf - iou_gt) - basemask * conf * conf);
            // class NLL over the batch-tiled class dim (CS=20)
            const int bs_idx = b / CSC, cs_idx = b - bs_idx * CSC;
            const size_t posoff = (size_t)(bn * 6 + 5) * HW + (size_t)(gj * NW + gi);
            float L[CSC];
            float m = -3.4e38f;
            #pragma unroll
            for (int c = 0; c < CSC; ++c) {
                L[c] = out[(size_t)(bs_idx * CSC + c) * 30 * HW + posoff];
                m = fmaxf(m, L[c]);
            }
            float se = 0.0f;
            #pragma unroll
            for (int c = 0; c < CSC; ++c) se += expf(L[c] - m);
            const float lse = m + logf(se);
            delta += -(L[cs_idx] - lse);   // CLASS_SCALE = 1
        }
    }
    sred[tid] = delta;
    __syncthreads();
    for (int s = 32; s > 0; s >>= 1) {
        if (tid < s) sred[tid] += sred[tid + s];
        __syncthreads();
    }
    if (tid == 0) P2[b] = sred[0];
}

// ---------------------------------------------------------------------------
// Kernel 3a: class means (enews, zero-padded to 32 rows for WMMA), counts,
// and `same` residual. 1024 threads, one per embedding dimension.
// ---------------------------------------------------------------------------
__global__ __launch_bounds__(1024) void k_meta1(const float* __restrict__ dw,
                                                const int* __restrict__ cls,
                                                float* __restrict__ enews,
                                                float* __restrict__ cnt_out,
                                                float* __restrict__ same_out) {
    __shared__ float scnt[NMETA];
    __shared__ int   scls[DW_N];
    __shared__ float sred[1024];
    const int tid = threadIdx.x;
    if (tid < NMETA) scnt[tid] = 0.0f;
    __syncthreads();
    if (tid < DW_N) {
        scls[tid] = cls[tid];
        atomicAdd(&scnt[cls[tid]], 1.0f);
    }
    __syncthreads();

    const int d = tid;
    float e_local[NMETA];
    #pragma unroll 4
    for (int c = 0; c < NMETA; ++c) {
        float s = 0.0f;
        for (int n = 0; n < DW_N; ++n)
            if (scls[n] == c) s += dw[n * DW_D + d];
        float e = s / fmaxf(scnt[c], 1.0f);
        e_local[c] = e;
        enews[c * DW_D + d] = e;
    }
    #pragma unroll
    for (int c = NMETA; c < EPAD; ++c) enews[c * DW_D + d] = 0.0f;  // zero padding rows

    float same = 0.0f;
    for (int n = 0; n < DW_N; ++n) {
        float v = dw[n * DW_D + d] - e_local[scls[n]];
        same += v * v;
    }
    sred[tid] = same;
    __syncthreads();
    for (int s = 512; s > 0; s >>= 1) {
        if (tid < s) sred[tid] += sred[tid + s];
        __syncthreads();
    }
    if (tid == 0) same_out[0] = sred[0];
    if (tid < NMETA) cnt_out[tid] = scnt[tid];
}

// ---------------------------------------------------------------------------
// Kernel 3b: Gram matrix G = E * E^T (32x1024, rows >=20 are zero) with
// v_wmma_f32_16x16x4_f32; d[i][j] = G[ii]+G[jj]-2G[ij]; dmin -> `different`.
// 4 waves, one 16x16 tile each; unconditional loads (no EXEC masking in the
// hot loop thanks to the zero padding). The distance matrix is symmetric, so
// the result is invariant to a transposed B-operand lane layout.
// ---------------------------------------------------------------------------
__global__ __launch_bounds__(128) void k_meta2(const float* __restrict__ enews,
                                               const float* __restrict__ cnt,
                                               float* __restrict__ diff_out) {
    __shared__ float sG[32 * 32];
    __shared__ float sdmin[NMETA];
    const int tid  = threadIdx.x;
    const int wave = tid >> 5;
    const int lane = tid & 31;
    const int ti   = wave >> 1;
    const int tj   = wave & 1;
    const int m    = lane & 15;
    const int koff = (lane >> 4) * 2;   // 16x4 f32 A layout: lanes 16-31 hold K=2,3
    const float* pa = enews + (size_t)(ti * 16 + m) * DW_D + koff;
    const float* pb = enews + (size_t)(tj * 16 + m) * DW_D + koff;

    v8f acc = {};
    #pragma unroll 4
    for (int k0 = 0; k0 < DW_D; k0 += 4) {
        v2f a, bb;
        a.x  = pa[k0];
        a.y  = pa[k0 + 1];
        bb.x = pb[k0];
        bb.y = pb[k0 + 1];
        acc = __builtin_amdgcn_wmma_f32_16x16x4_f32(
            /*neg_a=*/false, a, /*neg_b=*/false, bb,
            /*c_mod=*/(short)0, acc, /*reuse_a=*/false, /*reuse_b=*/false);
    }

    // C/D layout: VGPR r -> M=r (lanes 0-15), M=r+8 (lanes 16-31); N=lane&15
    const int colN   = lane & 15;
    const int rowOfs = (lane >> 4) * 8;
    #pragma unroll
    for (int r = 0; r < 8; ++r) {
        const int row = ti * 16 + rowOfs + r;
        const int col = tj * 16 + colN;
        sG[row * 32 + col] = acc[r];
    }
    __syncthreads();

    if (tid < NMETA) {
        const bool pi = cnt[tid] > 0.0f;
        float dmin = INFINITY;
        bool  any  = false;
        for (int j = 0; j < NMETA; ++j) {
            if (j == tid) continue;
            const bool pj = cnt[j] > 0.0f;
            if (pi && pj) {
                float dij = sG[tid * 32 + tid] + sG[j * 32 + j] - 2.0f * sG[tid * 32 + j];
                any  = true;
                dmin = fminf(dmin, dij);
            }
        }
        sdmin[tid] = any ? dmin : 0.0f;
    }
    __syncthreads();
    if (tid == 0) {
        float s = 0.0f;
        for (int j = 0; j < NMETA; ++j) s += sdmin[j];
        diff_out[0] = s;
    }
}

// ---------------------------------------------------------------------------
// Kernel 4: deterministic final reduction.
// ---------------------------------------------------------------------------
__global__ __launch_bounds__(256) void k_final(const float* __restrict__ P1, int n1,
                                               const float* __restrict__ P2, int n2,
                                               const float* __restrict__ same_v,
                                               const float* __restrict__ diff_v,
                                               float* __restrict__ outp) {
    __shared__ float sred[256];
    const int tid = threadIdx.x;
    float s = 0.0f;
    for (int i = tid; i < n1; i += 256) s += P1[i];
    for (int i = tid; i < n2; i += 256) s += P2[i];
    sred[tid] = s;
    __syncthreads();
    for (int st = 128; st > 0; st >>= 1) {
        if (tid < st) sred[tid] += sred[tid + st];
        __syncthreads();
    }
    if (tid == 0) outp[0] = sred[0] + same_v[0] - diff_v[0];
}

extern "C" void kernel_launch(void* const* d_in, const int* in_sizes, int n_in,
                              void* d_out, int out_size, void* d_ws, size_t ws_size,
                              hipStream_t stream) {
    const float* out = (const float*)d_in[0];   // (320,30,26,26)
    const float* tgt = (const float*)d_in[1];   // (16,20,250) == (320,50,5)
    const float* dw  = (const float*)d_in[2];   // (1,64,1024)
    const int*   cls = (const int*)d_in[3];     // (64,)

    float* ws     = (float*)d_ws;
    float* P1     = ws;            // 4480 = 320 * 14
    float* P2     = ws + 4480;     // 320
    float* same_v = ws + 4800;     // 1
    float* diff_v = ws + 4801;     // 1
    float* cnt    = ws + 4808;     // 20
    float* enews  = ws + 4832;     // 32*1024 (rows 20..31 zero)

    const int KT = (KPOS + 255) / 256;   // 14
    dim3 g1(NB, KT);
    k_base<<<g1, 256, 0, stream>>>(out, tgt, P1);
    k_scatter<<<NB, 64, 0, stream>>>(out, tgt, P2);
    k_meta1<<<1, 1024, 0, stream>>>(dw, cls, enews, cnt, same_v);
    k_meta2<<<1, 128, 0, stream>>>(enews, cnt, diff_v);
    k_final<<<1, 256, 0, stream>>>(P1, NB * KT, P2, NB, same_v, diff_v, (float*)d_out);
}